// depth_channel_att_44332652430130
// MI455X (gfx1250) — compile-verified
//
#include <hip/hip_runtime.h>
#include <math.h>

// Problem constants (from reference): B=32, C=256, H=W=64, K=3
#define B_    32
#define C_    256
#define HW_   4096          // 64*64
#define NOUT_ 768           // K*C
#define BN_EPS_ 1e-5f

typedef float v2f __attribute__((ext_vector_type(2)));
typedef float v8f __attribute__((ext_vector_type(8)));

// ---------------------------------------------------------------------------
// Kernel 1: global average pool over H*W for each (b,c) plane.
// grid = B*C blocks, 256 threads. Each block reduces 4096 floats (16 KB).
// ---------------------------------------------------------------------------
__global__ void gap_kernel(const float* __restrict__ x, float* __restrict__ g)
{
    const int plane = blockIdx.x;                      // b*C + c
    const float4* xp = (const float4*)(x + (size_t)plane * HW_);
    float s = 0.f;
    // 4096 floats = 1024 float4; 256 threads -> 4 float4 each
#pragma unroll
    for (int i = 0; i < 4; ++i) {
        float4 v = xp[threadIdx.x + i * 256];
        s += v.x + v.y + v.z + v.w;
    }
    // wave32 butterfly reduce
#pragma unroll
    for (int off = 16; off > 0; off >>= 1)
        s += __shfl_xor(s, off, 32);
    __shared__ float lds[8];
    const int wid  = threadIdx.x >> 5;
    const int lane = threadIdx.x & 31;
    if (lane == 0) lds[wid] = s;
    __syncthreads();
    if (threadIdx.x == 0) {
        float t = 0.f;
#pragma unroll
        for (int w = 0; w < 8; ++w) t += lds[w];
        g[plane] = t * (1.0f / (float)HW_);
    }
}

// ---------------------------------------------------------------------------
// Kernel 2: y = g (32x256) * conv_w^T (256x768) via V_WMMA_F32_16X16X4_F32,
// fused BN (inference stats) + tanh epilogue -> f (32x768).
// One wave per 16x16 output tile: 2 M-tiles * 48 N-tiles = 96 waves.
// blockDim must be exactly 32 (EXEC all ones, as WMMA requires).
//
// VGPR layouts per CDNA5 ISA 7.12.2:
//   A 16x4 f32 : lane L holds A[M = L%16, K = 2*(L/16) + j] in component j
//   B 4x16 f32 : lane L holds B[K = 2*(L/16) + j, N = L%16] in component j
//   C/D 16x16  : VGPR r, lanes 0-15 -> (M=r, N=lane); lanes 16-31 -> (M=r+8)
// ---------------------------------------------------------------------------
__global__ void filt_gemm_wmma_kernel(const float* __restrict__ g,    // (32,256)
                                      const float* __restrict__ w,    // (768,256)
                                      const float* __restrict__ bnw,
                                      const float* __restrict__ bnb,
                                      const float* __restrict__ bnm,
                                      const float* __restrict__ bnv,
                                      float* __restrict__ f)          // (32,768)
{
    const int tile = blockIdx.x;        // 0..95
    const int mt   = tile & 1;          // m-tile: 0..1
    const int nt   = tile >> 1;         // n-tile: 0..47
    const int lane = threadIdx.x;       // 0..31
    const int ln   = lane & 15;
    const int lh   = lane >> 4;         // 0/1: selects K pair (A/B) and M half (C/D)

    const float* arow = g + (size_t)(mt * 16 + ln) * C_;   // A row  (g[m, :])
    const float* brow = w + (size_t)(nt * 16 + ln) * C_;   // conv_w[n, :] = B[:, n]

    v8f acc = {};
    for (int k0 = 0; k0 < C_; k0 += 4) {
        v2f a, b;
        a.x = arow[k0 + 2 * lh + 0];
        a.y = arow[k0 + 2 * lh + 1];
        b.x = brow[k0 + 2 * lh + 0];
        b.y = brow[k0 + 2 * lh + 1];
        // 8 args: (neg_a, A, neg_b, B, c_mod, C, reuse_a, reuse_b)
        acc = __builtin_amdgcn_wmma_f32_16x16x4_f32(
            false, a, false, b, (short)0, acc, false, false);
    }

    // Epilogue: yn = y*scale + bias;  f = tanh(yn)
    const int   n     = nt * 16 + ln;
    const float scale = bnw[n] * rsqrtf(bnv[n] + BN_EPS_);
    const float bias  = bnb[n] - bnm[n] * scale;
#pragma unroll
    for (int r = 0; r < 8; ++r) {
        const int m = mt * 16 + lh * 8 + r;
        f[(size_t)m * NOUT_ + n] = tanhf(acc[r] * scale + bias);
    }
}

// ---------------------------------------------------------------------------
// Kernel 3: out[b,c,:] = gamma[c] * ( f0*x[b,refl(c-1)] + f1*x[b,c]
//                                   + f2*x[b,refl(c+1)] ) + beta[c]*x[b,c]
// Reflect pad along channels: refl(-1)=1, refl(256)=254.
// grid = (C, B), 256 threads, float4 streaming (memory bound; neighbor
// channel planes are L2 hits since all of x fits in the 192 MB L2).
// ---------------------------------------------------------------------------
__global__ void apply_kernel(const float* __restrict__ x,
                             const float* __restrict__ f,      // (32,768)
                             const float* __restrict__ gamma,  // (256)
                             const float* __restrict__ beta,   // (256)
                             float* __restrict__ out)
{
    const int c = blockIdx.x;           // 0..255
    const int b = blockIdx.y;           // 0..31
    const int cm = (c == 0)      ? 1        : c - 1;
    const int cp = (c == C_ - 1) ? C_ - 2   : c + 1;

    const float f0 = f[(size_t)b * NOUT_ + 0 * C_ + c];
    const float f1 = f[(size_t)b * NOUT_ + 1 * C_ + c];
    const float f2 = f[(size_t)b * NOUT_ + 2 * C_ + c];
    const float ga = gamma[c];
    const float be = beta[c];

    const float4* xm = (const float4*)(x   + ((size_t)b * C_ + cm) * HW_);
    const float4* xc = (const float4*)(x   + ((size_t)b * C_ + c ) * HW_);
    const float4* xp = (const float4*)(x   + ((size_t)b * C_ + cp) * HW_);
    float4*       op = (float4*)      (out + ((size_t)b * C_ + c ) * HW_);

#pragma unroll
    for (int i = 0; i < 4; ++i) {
        const int idx = threadIdx.x + i * 256;
        const float4 a = xm[idx];
        const float4 m = xc[idx];
        const float4 p = xp[idx];
        float4 o;
        o.x = ga * (f0 * a.x + f1 * m.x + f2 * p.x) + be * m.x;
        o.y = ga * (f0 * a.y + f1 * m.y + f2 * p.y) + be * m.y;
        o.z = ga * (f0 * a.z + f1 * m.z + f2 * p.z) + be * m.z;
        o.w = ga * (f0 * a.w + f1 * m.w + f2 * p.w) + be * m.w;
        op[idx] = o;
    }
}

// ---------------------------------------------------------------------------
// Launch: inputs (in setup_inputs order):
//   0: x (B,C,H,W) f32   1: conv_w (768,256) f32
//   2: bn_weight  3: bn_bias  4: bn_mean  5: bn_var   (768 each)
//   6: gamma (256)  7: beta (256)
// Workspace: g at ws+0 (32*256 f32 = 32 KB), f at ws+32KB (32*768 f32 = 96 KB)
// ---------------------------------------------------------------------------
extern "C" void kernel_launch(void* const* d_in, const int* in_sizes, int n_in,
                              void* d_out, int out_size, void* d_ws, size_t ws_size,
                              hipStream_t stream)
{
    const float* x      = (const float*)d_in[0];
    const float* conv_w = (const float*)d_in[1];
    const float* bnw    = (const float*)d_in[2];
    const float* bnb    = (const float*)d_in[3];
    const float* bnm    = (const float*)d_in[4];
    const float* bnv    = (const float*)d_in[5];
    const float* gamma  = (const float*)d_in[6];
    const float* beta   = (const float*)d_in[7];
    float*       out    = (float*)d_out;

    float* g_ws = (float*)d_ws;                           // 32*256
    float* f_ws = (float*)((char*)d_ws + B_ * C_ * sizeof(float)); // 32*768

    gap_kernel<<<B_ * C_, 256, 0, stream>>>(x, g_ws);
    filt_gemm_wmma_kernel<<<96, 32, 0, stream>>>(g_ws, conv_w, bnw, bnb, bnm, bnv, f_ws);
    apply_kernel<<<dim3(C_, B_), 256, 0, stream>>>(x, f_ws, gamma, beta, out);
}